// RotationGate_34583076667988
// MI455X (gfx1250) — compile-verified
//
#include <hip/hip_runtime.h>

// RX gate on qubit 10 of a 22-qubit batched state (batch=4), mapped onto
// v_wmma_f32_16x16x4_f32. Memory-bound (268 MB @ 23.3 TB/s ~= 11.5 us floor);
// WMMA does the per-site 16x16 (batch-block-diagonal) real contraction for
// free while loads/stores stay fully coalesced.

typedef __attribute__((ext_vector_type(2))) float v2f;
typedef __attribute__((ext_vector_type(4))) float f4;
typedef __attribute__((ext_vector_type(8))) float v8f;

static constexpr int N_QUBITS = 22;
static constexpr int QUBIT    = 10;
static constexpr int BATCH    = 4;
static constexpr int A_DIM    = 1 << QUBIT;                    // 1024
static constexpr int C_DIM    = 1 << (N_QUBITS - QUBIT - 1);   // 2048
static constexpr int NSITES   = A_DIM * C_DIM;                 // 2^21 (a,c) sites
static constexpr int NGROUPS  = NSITES / 16;                   // 16 sites per WMMA

__global__ void __launch_bounds__(256)
rx_gate_wmma_kernel(const float* __restrict__ theta,
                    const float* __restrict__ sre,
                    const float* __restrict__ sim,
                    const float* __restrict__ P,
                    float* __restrict__ out)
{
    const int lane = threadIdx.x & 31;
    const int m    = lane & 15;   // row of G (A) / site column (B,D)
    const int hi   = lane >> 4;   // half-wave: selects K half (A,B) and M half (D)
    const int gm   = m >> 2;      // output group: 0=re0 1=im0 2=re1 3=im1
    const int bm   = m & 3;       // batch index of this G row

    // Per-lane rotation coefficients (only batch bm is ever non-zero in row m).
    const float th = theta[bm] * 0.5f;
    const float cb = __cosf(th) * 0.0f + cosf(th);
    const float sb = sinf(th);
    const float p0 = P[(gm >> 1) * 2 + 0];   // P[gm/2][0]
    const float p1 = P[(gm >> 1) * 2 + 1];   // P[gm/2][1]

    // A fragments for the 4 K-chunks (gk = r). Column k = 4r + (2*hi + t):
    // non-zero only when its batch (2*hi+t) == bm. Branchless -> EXEC stays ~0.
    v2f afrag[4];
#pragma unroll
    for (int r = 0; r < 4; ++r) {
        const float prow  = ((r >> 1) == 0) ? p0 : p1;              // P[gm/2][r/2]
        const float sterm = ((gm & 1) == 0) ? (sb * prow) : (-sb * prow);
        const float gv    = ((gm == r) ? cb : 0.0f)
                          + ((((gm ^ r) & 1) != 0) ? sterm : 0.0f);
        afrag[r][0] = (bm == 2 * hi + 0) ? gv : 0.0f;
        afrag[r][1] = (bm == 2 * hi + 1) ? gv : 0.0f;
    }

    const int wavesPerBlock = blockDim.x >> 5;
    const int gid     = blockIdx.x * wavesPerBlock + (threadIdx.x >> 5);
    const int gstride = gridDim.x * wavesPerBlock;

    for (int g = gid; g < NGROUPS; g += gstride) {   // wave-uniform loop
        const int s0 = g << 4;                       // first site of group
        const int a  = s0 >> 11;                     // / C_DIM
        const int c  = (s0 & (C_DIM - 1)) + m;       // 16 | C_DIM, no a-crossing

        // float index of (a, j, c, batch 2*hi) in re/im buffers
        const size_t row0 = ((size_t)(2 * a + 0) * C_DIM + c) * BATCH + 2 * hi;
        const size_t row1 = ((size_t)(2 * a + 1) * C_DIM + c) * BATCH + 2 * hi;

        // B fragments: chunk r covers w-rows 4r..4r+3 = {re0,im0,re1,im1}[b0..3]
        const v2f b0 = *(const v2f*)(sre + row0);
        const v2f b1 = *(const v2f*)(sim + row0);
        const v2f b2 = *(const v2f*)(sre + row1);
        const v2f b3 = *(const v2f*)(sim + row1);

        v8f acc = {};
        acc = __builtin_amdgcn_wmma_f32_16x16x4_f32(false, afrag[0], false, b0,
                                                    (short)0, acc, false, false);
        acc = __builtin_amdgcn_wmma_f32_16x16x4_f32(false, afrag[1], false, b1,
                                                    (short)0, acc, false, false);
        acc = __builtin_amdgcn_wmma_f32_16x16x4_f32(false, afrag[2], false, b2,
                                                    (short)0, acc, false, false);
        acc = __builtin_amdgcn_wmma_f32_16x16x4_f32(false, afrag[3], false, b3,
                                                    (short)0, acc, false, false);

        // D: lane (hi=0) holds rows 0..7 = (re_j0[b], im_j0[b]) of site m,
        //    lane (hi=1) holds rows 8..15 = j=1 outputs. Interleave re/im pairs
        //    and store complex64 as two contiguous b128 per lane.
        const size_t ob = ((size_t)(2 * a + hi) * C_DIM + c) * (BATCH * 2);
        f4 o0 = {acc[0], acc[4], acc[1], acc[5]};
        f4 o1 = {acc[2], acc[6], acc[3], acc[7]};
        *(f4*)(out + ob + 0) = o0;
        *(f4*)(out + ob + 4) = o1;
    }
}

extern "C" void kernel_launch(void* const* d_in, const int* in_sizes, int n_in,
                              void* d_out, int out_size, void* d_ws, size_t ws_size,
                              hipStream_t stream) {
    const float* theta = (const float*)d_in[0];
    const float* sre   = (const float*)d_in[1];
    const float* sim   = (const float*)d_in[2];
    const float* P     = (const float*)d_in[3];
    float* out = (float*)d_out;   // complex64 -> interleaved float pairs

    (void)in_sizes; (void)n_in; (void)out_size; (void)d_ws; (void)ws_size;

    // 4096 blocks x 8 wave32 = 32768 waves; 131072 groups -> 4 groups/wave.
    rx_gate_wmma_kernel<<<4096, 256, 0, stream>>>(theta, sre, sim, P, out);
}